// GATv2Encoder_70300024701697
// MI455X (gfx1250) — compile-verified
//
#include <hip/hip_runtime.h>
#include <math.h>

typedef __attribute__((ext_vector_type(2))) float v2f;
typedef __attribute__((ext_vector_type(8))) float v8f;

#define NEG_SLOPE 0.2f

// ---------- helpers ----------
__device__ __forceinline__ unsigned f2ord(float f) {
    unsigned u = __float_as_uint(f);
    return (u & 0x80000000u) ? ~u : (u | 0x80000000u);
}
__device__ __forceinline__ float ord2f(unsigned u) {
    return (u & 0x80000000u) ? __uint_as_float(u & 0x7fffffffu)
                             : __uint_as_float(~u);
}
__device__ __forceinline__ float lrelu(float m) {
    return (m > 0.f) ? m : NEG_SLOPE * m;
}

// ---------- WMMA f32 GEMM:  C[M,Nout] = A[M,K] @ W[K,Nout] (+bias) ----------
// block = 128 threads (4 waves); wave w handles a 16-col subtile.
// grid = (ceil(M/16), Nout/64).  Two independent accumulator chains so two
// v_wmma_f32_16x16x4_f32 can be in flight per wave.
__global__ void k_gemm_wmma(const float* __restrict__ A,
                            const float* __restrict__ W,
                            const float* __restrict__ bias,
                            float* __restrict__ C,
                            int M, int K, int Nout) {
    const int wave  = threadIdx.x >> 5;
    const int lane  = threadIdx.x & 31;
    const int r     = lane & 15;       // lane within half-wave
    const int sel   = lane >> 4;       // 0: K pair {0,1}, 1: K pair {2,3}
    const int tileM = blockIdx.x * 16;
    const int tileN = blockIdx.y * 64 + wave * 16;

    int rowA = tileM + r;
    int rowC = rowA < M ? rowA : (M - 1);     // clamp for tail tiles
    const float* Arow = A + (size_t)rowC * K;

    v8f acc0 = {}, acc1 = {};
    for (int k = 0; k < K; k += 8) {          // K is a multiple of 8 (128/256)
        const int ka = k + 2 * sel;
        v2f a0, b0, a1, b1;
        a0.x = Arow[ka];
        a0.y = Arow[ka + 1];
        a1.x = Arow[ka + 4];
        a1.y = Arow[ka + 5];
        b0.x = W[(size_t)ka * Nout + tileN + r];
        b0.y = W[(size_t)(ka + 1) * Nout + tileN + r];
        b1.x = W[(size_t)(ka + 4) * Nout + tileN + r];
        b1.y = W[(size_t)(ka + 5) * Nout + tileN + r];
        acc0 = __builtin_amdgcn_wmma_f32_16x16x4_f32(
            false, a0, false, b0, (short)0, acc0, false, false);
        acc1 = __builtin_amdgcn_wmma_f32_16x16x4_f32(
            false, a1, false, b1, (short)0, acc1, false, false);
    }

    const float bv = bias ? bias[tileN + r] : 0.0f;
#pragma unroll
    for (int v = 0; v < 8; ++v) {
        const int row = tileM + v + 8 * sel;   // D layout: VGPR v -> row v / v+8
        if (row < M)
            C[(size_t)row * Nout + tileN + r] = acc0[v] + acc1[v] + bv;
    }
}

// ---------- logits + segment max.  One wave per edge (or node for self-loops).
// HC = H*C (256 or 512); 8 lanes per head in both layers. ----------
template <int HC>
__global__ void k_logits(const float* __restrict__ xl,
                         const float* __restrict__ xr,
                         const float* __restrict__ rel_e,   // null => self-loop (e=0)
                         const int* __restrict__ src,       // null => self-loop
                         const int* __restrict__ rel,
                         const int* __restrict__ dst,
                         const float* __restrict__ att,     // [HC]
                         float* __restrict__ logits,        // [count,4]
                         unsigned* __restrict__ maxbuf,     // [N,4] ordered-uint
                         int count) {
    const int VPL  = HC / 32;          // 8 or 16 floats per lane
    const int NV4  = VPL / 4;          // 2 or 4 float4 per lane
    const int wave = (int)((blockIdx.x * (size_t)blockDim.x + threadIdx.x) >> 5);
    const int lane = threadIdx.x & 31;
    if (wave >= count) return;

    const int s = src ? src[wave] : wave;
    const int d = dst ? dst[wave] : wave;

    const float4* ps = (const float4*)(xl + (size_t)s * HC + lane * VPL);
    const float4* pd = (const float4*)(xr + (size_t)d * HC + lane * VPL);
    const float4* pa = (const float4*)(att + lane * VPL);
    const float4* pe = rel_e
        ? (const float4*)(rel_e + (size_t)rel[wave] * HC + lane * VPL) : nullptr;

    float sum = 0.f;
#pragma unroll
    for (int j = 0; j < NV4; ++j) {
        const float4 vs = ps[j];
        const float4 vd = pd[j];
        const float4 va = pa[j];
        const float4 ve = pe ? pe[j] : make_float4(0.f, 0.f, 0.f, 0.f);
        sum += lrelu(vs.x + vd.x + ve.x) * va.x;
        sum += lrelu(vs.y + vd.y + ve.y) * va.y;
        sum += lrelu(vs.z + vd.z + ve.z) * va.z;
        sum += lrelu(vs.w + vd.w + ve.w) * va.w;
    }
    // reduce over 8 lanes (one head per 8-lane group)
    sum += __shfl_xor(sum, 1, 32);
    sum += __shfl_xor(sum, 2, 32);
    sum += __shfl_xor(sum, 4, 32);

    if ((lane & 7) == 0) {
        const int h = lane >> 3;
        logits[(size_t)wave * 4 + h] = sum;
        atomicMax(&maxbuf[(size_t)d * 4 + h], f2ord(sum));
    }
}

// ---------- a = exp(logit - max[dst]); denom[dst] += a. One thread per edge/node.
__global__ void k_expsum(float* __restrict__ logits,        // in/out [count,4]
                         const int* __restrict__ dst,       // null => self
                         const unsigned* __restrict__ maxbuf,
                         float* __restrict__ denom,
                         int count) {
    const int e = (int)(blockIdx.x * (size_t)blockDim.x + threadIdx.x);
    if (e >= count) return;
    const int d = dst ? dst[e] : e;
    float4 lg = *(float4*)(logits + (size_t)e * 4);
    const uint4 mo = *(const uint4*)(maxbuf + (size_t)d * 4);
    lg.x = __expf(lg.x - ord2f(mo.x));
    lg.y = __expf(lg.y - ord2f(mo.y));
    lg.z = __expf(lg.z - ord2f(mo.z));
    lg.w = __expf(lg.w - ord2f(mo.w));
    *(float4*)(logits + (size_t)e * 4) = lg;
    float* dn = denom + (size_t)d * 4;
    atomicAdd(&dn[0], lg.x);
    atomicAdd(&dn[1], lg.y);
    atomicAdd(&dn[2], lg.z);
    atomicAdd(&dn[3], lg.w);
}

// ---------- acc[dst] += (a/denom[dst]) * xl[src]. One wave per edge/node. ----------
template <int HC>
__global__ void k_aggr(const float* __restrict__ xl,
                       const float* __restrict__ a,         // [count,4]
                       const int* __restrict__ src,         // null => self
                       const int* __restrict__ dst,
                       const float* __restrict__ denom,
                       float* __restrict__ acc,
                       int count) {
    const int VPL  = HC / 32;
    const int NV4  = VPL / 4;
    const int wave = (int)((blockIdx.x * (size_t)blockDim.x + threadIdx.x) >> 5);
    const int lane = threadIdx.x & 31;
    if (wave >= count) return;

    const int s = src ? src[wave] : wave;
    const int d = dst ? dst[wave] : wave;
    const int h = lane >> 3;
    const float alpha = a[(size_t)wave * 4 + h] / denom[(size_t)d * 4 + h];

    const float4* px = (const float4*)(xl + (size_t)s * HC + lane * VPL);
    float* pacc      = acc + (size_t)d * HC + lane * VPL;
#pragma unroll
    for (int j = 0; j < NV4; ++j) {
        const float4 v = px[j];
        atomicAdd(&pacc[j * 4 + 0], alpha * v.x);
        atomicAdd(&pacc[j * 4 + 1], alpha * v.y);
        atomicAdd(&pacc[j * 4 + 2], alpha * v.z);
        atomicAdd(&pacc[j * 4 + 3], alpha * v.w);
    }
}

// ---------- finalize layer1: h = acc + bias (vectorized, Nout=256 pow2) ----------
__global__ void k_add_bias(const float* __restrict__ acc,
                           const float* __restrict__ bias,
                           float* __restrict__ out, int total4, int mask) {
    const int i4 = (int)(blockIdx.x * (size_t)blockDim.x + threadIdx.x);
    if (i4 >= total4) return;
    const int idx = i4 * 4;
    const float4 v = *(const float4*)(acc + idx);
    const float4 b = *(const float4*)(bias + (idx & mask));
    float4 o;
    o.x = v.x + b.x; o.y = v.y + b.y; o.z = v.z + b.z; o.w = v.w + b.w;
    *(float4*)(out + idx) = o;
}

// ---------- finalize layer2: mean over 4 heads (+bias), OUT=128 ----------
__global__ void k_mean_heads(const float* __restrict__ acc,   // [N,512]
                             const float* __restrict__ bias,  // [128]
                             float* __restrict__ out, int total4) {
    const int i4 = (int)(blockIdx.x * (size_t)blockDim.x + threadIdx.x);
    if (i4 >= total4) return;                 // total4 = N*128/4
    const int n = i4 >> 5;                    // 32 float4 per node
    const int o = (i4 & 31) * 4;
    const float* p = acc + (size_t)n * 512 + o;
    const float4 v0 = *(const float4*)(p);
    const float4 v1 = *(const float4*)(p + 128);
    const float4 v2 = *(const float4*)(p + 256);
    const float4 v3 = *(const float4*)(p + 384);
    const float4 b  = *(const float4*)(bias + o);
    float4 r;
    r.x = 0.25f * (v0.x + v1.x + v2.x + v3.x) + b.x;
    r.y = 0.25f * (v0.y + v1.y + v2.y + v3.y) + b.y;
    r.z = 0.25f * (v0.z + v1.z + v2.z + v3.z) + b.z;
    r.w = 0.25f * (v0.w + v1.w + v2.w + v3.w) + b.w;
    *(float4*)(out + (size_t)n * 128 + o) = r;
}

// =====================================================================
extern "C" void kernel_launch(void* const* d_in, const int* in_sizes, int n_in,
                              void* d_out, int out_size, void* d_ws, size_t ws_size,
                              hipStream_t stream) {
    const float* x    = (const float*)d_in[0];
    const int*   ei   = (const int*)d_in[1];
    const float* rels = (const float*)d_in[2];
    const float* Wl1  = (const float*)d_in[3];
    const float* bl1  = (const float*)d_in[4];
    const float* Wr1  = (const float*)d_in[5];
    const float* br1  = (const float*)d_in[6];
    const float* We1  = (const float*)d_in[7];
    const float* att1 = (const float*)d_in[8];
    const float* bias1= (const float*)d_in[9];
    const float* Wl2  = (const float*)d_in[10];
    const float* bl2  = (const float*)d_in[11];
    const float* Wr2  = (const float*)d_in[12];
    const float* br2  = (const float*)d_in[13];
    const float* We2  = (const float*)d_in[14];
    const float* att2 = (const float*)d_in[15];
    const float* bias2= (const float*)d_in[16];
    float* out = (float*)d_out;

    const int N = in_sizes[0] / 128;      // 50000
    const int E = in_sizes[1] / 3;        // 400000
    const int R = in_sizes[2] / 128;      // 500
    const int* srcI = ei;
    const int* relI = ei + E;
    const int* dstI = ei + 2 * E;

    // ---- workspace layout (reuse: buf1+buf2 -> xl2, buf3+buf5 -> xr2) ----
    float* p = (float*)d_ws;
    const size_t nb = (size_t)N * 256;
    float* buf1 = p; p += nb;             // xl1, later xl2 (low half)
    float* buf2 = p; p += nb;             // xr1, later xl2 (high half)
    float* buf3 = p; p += nb;             // acc1, later xr2 (low half)
    float* buf5 = p; p += nb;             // xr2 (high half)
    float* buf4 = p; p += nb;             // h (layer-1 output)
    float* acc2 = p; p += (size_t)N * 512;
    float* rel1 = p; p += (size_t)R * 256;
    float* rel2 = p; p += (size_t)R * 512;
    float* log1 = p; p += (size_t)E * 4;
    float* log2 = p; p += (size_t)E * 4;
    float* slf1 = p; p += (size_t)N * 4;
    float* slf2 = p; p += (size_t)N * 4;
    float* den1 = p; p += (size_t)N * 4;
    float* den2 = p; p += (size_t)N * 4;
    unsigned* max1 = (unsigned*)p; p += (size_t)N * 4;
    unsigned* max2 = (unsigned*)p; p += (size_t)N * 4;
    float* xl2 = buf1;                    // contiguous N*512
    float* xr2 = buf3;                    // contiguous N*512

    // ---- zero-init accumulators (graph-capture safe) ----
    hipMemsetAsync(buf3, 0, nb * sizeof(float), stream);               // acc1
    hipMemsetAsync(acc2, 0, (size_t)N * 512 * sizeof(float), stream);
    hipMemsetAsync(den1, 0, (size_t)N * 4 * sizeof(float), stream);
    hipMemsetAsync(den2, 0, (size_t)N * 4 * sizeof(float), stream);
    hipMemsetAsync(max1, 0, (size_t)N * 4 * sizeof(unsigned), stream); // 0 < f2ord(-inf)
    hipMemsetAsync(max2, 0, (size_t)N * 4 * sizeof(unsigned), stream);

    const dim3 gemmBlk(128);
    const int mTilesN = (N + 15) / 16;    // 3125 (exact)
    const int mTilesR = (R + 15) / 16;    // 32

    // ---- per-relation edge embeddings (tiny GEMMs instead of E-wide) ----
    k_gemm_wmma<<<dim3(mTilesR, 256 / 64), gemmBlk, 0, stream>>>(rels, We1, nullptr, rel1, R, 128, 256);
    k_gemm_wmma<<<dim3(mTilesR, 512 / 64), gemmBlk, 0, stream>>>(rels, We2, nullptr, rel2, R, 128, 512);

    // ---- layer 1 node transforms ----
    k_gemm_wmma<<<dim3(mTilesN, 256 / 64), gemmBlk, 0, stream>>>(x, Wl1, bl1, buf1, N, 128, 256);
    k_gemm_wmma<<<dim3(mTilesN, 256 / 64), gemmBlk, 0, stream>>>(x, Wr1, br1, buf2, N, 128, 256);

    const int edgeBlocks = (E + 7) / 8;   // 8 waves / 256-thread block
    const int nodeBlocks = (N + 7) / 8;

    // ---- layer 1 attention ----
    k_logits<256><<<edgeBlocks, 256, 0, stream>>>(buf1, buf2, rel1, srcI, relI, dstI, att1, log1, max1, E);
    k_logits<256><<<nodeBlocks, 256, 0, stream>>>(buf1, buf2, nullptr, nullptr, nullptr, nullptr, att1, slf1, max1, N);
    k_expsum<<<(E + 255) / 256, 256, 0, stream>>>(log1, dstI, max1, den1, E);
    k_expsum<<<(N + 255) / 256, 256, 0, stream>>>(slf1, nullptr, max1, den1, N);
    k_aggr<256><<<edgeBlocks, 256, 0, stream>>>(buf1, log1, srcI, dstI, den1, buf3, E);
    k_aggr<256><<<nodeBlocks, 256, 0, stream>>>(buf1, slf1, nullptr, nullptr, den1, buf3, N);
    k_add_bias<<<(N * 256 / 4 + 255) / 256, 256, 0, stream>>>(buf3, bias1, buf4, N * 256 / 4, 255);

    // ---- layer 2 node transforms (input = h in buf4) ----
    k_gemm_wmma<<<dim3(mTilesN, 512 / 64), gemmBlk, 0, stream>>>(buf4, Wl2, bl2, xl2, N, 256, 512);
    k_gemm_wmma<<<dim3(mTilesN, 512 / 64), gemmBlk, 0, stream>>>(buf4, Wr2, br2, xr2, N, 256, 512);

    // ---- layer 2 attention ----
    k_logits<512><<<edgeBlocks, 256, 0, stream>>>(xl2, xr2, rel2, srcI, relI, dstI, att2, log2, max2, E);
    k_logits<512><<<nodeBlocks, 256, 0, stream>>>(xl2, xr2, nullptr, nullptr, nullptr, nullptr, att2, slf2, max2, N);
    k_expsum<<<(E + 255) / 256, 256, 0, stream>>>(log2, dstI, max2, den2, E);
    k_expsum<<<(N + 255) / 256, 256, 0, stream>>>(slf2, nullptr, max2, den2, N);
    k_aggr<512><<<edgeBlocks, 256, 0, stream>>>(xl2, log2, srcI, dstI, den2, acc2, E);
    k_aggr<512><<<nodeBlocks, 256, 0, stream>>>(xl2, slf2, nullptr, nullptr, den2, acc2, N);

    // ---- mean over heads + bias -> output ----
    k_mean_heads<<<(N * 128 / 4 + 255) / 256, 256, 0, stream>>>(acc2, bias2, out, N * 128 / 4);
}